// NeuronPooling_67954972557621
// MI455X (gfx1250) — compile-verified
//
#include <hip/hip_runtime.h>
#include <math.h>

// Problem constants (B,N,L,D) = (4,16,2048,512)
#define B_ 4
#define N_ 16
#define L_ 2048
#define D_ 512
#define RSQRT_D 0.044194173824159216f  // 1/sqrt(512)

typedef __attribute__((ext_vector_type(2))) float v2f;
typedef __attribute__((ext_vector_type(4))) float v4f;
typedef __attribute__((ext_vector_type(8))) float v8f;

// ---------------------------------------------------------------------------
// Kernel 0: M[t,d] = sum_i Wq[i,t] * Wk[i,d]   (M = Wq^T @ Wk, 512x512)
// One wave per 16x16 output tile, K-loop in steps of 4 via V_WMMA_F32_16X16X4_F32.
// f32 WMMA lane layouts (ISA 7.12.2):
//   A 16x4 : lane -> m = lane%16, VGPR v holds K = (lane/16)*2 + v
//   B 4x16 : lane -> n = lane%16, VGPR v holds K = (lane/16)*2 + v (mirror of A)
//   C/D    : lane -> col = lane%16, VGPR r holds row = r + (lane>=16 ? 8 : 0)
// ---------------------------------------------------------------------------
__global__ __launch_bounds__(128)
void combine_w_kernel(const float* __restrict__ Wq, const float* __restrict__ Wk,
                      float* __restrict__ M) {
    const int wave = (blockIdx.x * blockDim.x + threadIdx.x) >> 5;
    const int lane = threadIdx.x & 31;
    const int ti = wave >> 5;           // t-tile (32 tiles)
    const int tj = wave & 31;           // d-tile (32 tiles)
    const int tbase = ti * 16;
    const int dbase = tj * 16;
    const int mn = lane & 15;           // A row / B col within tile
    const int kb = (lane >> 4) << 1;    // K sub-offset per lane half

    v8f acc = {};
    for (int kk = 0; kk < D_; kk += 4) {
        v2f a, b;
        // A[m,k] = Wq^T[tbase+m, k] = Wq[k, tbase+m]
        a.x = Wq[(kk + kb    ) * D_ + tbase + mn];
        a.y = Wq[(kk + kb + 1) * D_ + tbase + mn];
        // B[k,n] = Wk[k, dbase+n]
        b.x = Wk[(kk + kb    ) * D_ + dbase + mn];
        b.y = Wk[(kk + kb + 1) * D_ + dbase + mn];
        acc = __builtin_amdgcn_wmma_f32_16x16x4_f32(
            false, a, false, b, (short)0, acc, false, false);
    }
    const int row0 = (lane >> 4) << 3;
    const int col  = lane & 15;
#pragma unroll
    for (int r = 0; r < 8; ++r)
        M[(tbase + row0 + r) * D_ + dbase + col] = acc[r];
}

// ---------------------------------------------------------------------------
// cvec[d] = sum_i bq[i] * Wk[i,d]   (tiny; one 512-thread block)
// ---------------------------------------------------------------------------
__global__ __launch_bounds__(512)
void bias_fold_kernel(const float* __restrict__ bq, const float* __restrict__ Wk,
                      float* __restrict__ cvec) {
    const int d = threadIdx.x;
    float s = 0.f;
    for (int i = 0; i < D_; ++i)
        s = fmaf(bq[i], Wk[i * D_ + d], s);
    cvec[d] = s;
}

// ---------------------------------------------------------------------------
// Kernel 1: qk = H @ M + cvec    (H = highway flattened to (B*L=8192, 512))
// One wave computes a 16x64 strip (4 accumulators) so the A fragment is
// reused 4x per K-step. H rows are contiguous in K -> single b64 load for A.
// ---------------------------------------------------------------------------
__global__ __launch_bounds__(128)
void qk_gemm_kernel(const float* __restrict__ H, const float* __restrict__ M,
                    const float* __restrict__ cvec, float* __restrict__ qk) {
    const int wave = (blockIdx.x * blockDim.x + threadIdx.x) >> 5;
    const int lane = threadIdx.x & 31;
    const int rtile  = wave >> 3;           // 512 row tiles
    const int cstrip = wave & 7;            // 8 column strips of 64
    const int rbase = rtile * 16;
    const int cbase = cstrip * 64;
    const int mn = lane & 15;
    const int kb = (lane >> 4) << 1;

    const float* arow = H + (size_t)(rbase + mn) * D_ + kb;

    v8f acc0 = {}, acc1 = {}, acc2 = {}, acc3 = {};
    for (int kk = 0; kk < D_; kk += 4) {
        v2f a = *(const v2f*)(arow + kk);   // H[row, kk+kb], H[row, kk+kb+1]
        const float* Mr0 = M + (size_t)(kk + kb) * D_ + cbase + mn;
        const float* Mr1 = Mr0 + D_;
        v2f b0, b1, b2, b3;
        b0.x = Mr0[ 0]; b0.y = Mr1[ 0];
        b1.x = Mr0[16]; b1.y = Mr1[16];
        b2.x = Mr0[32]; b2.y = Mr1[32];
        b3.x = Mr0[48]; b3.y = Mr1[48];
        acc0 = __builtin_amdgcn_wmma_f32_16x16x4_f32(false, a, false, b0, (short)0, acc0, false, false);
        acc1 = __builtin_amdgcn_wmma_f32_16x16x4_f32(false, a, false, b1, (short)0, acc1, false, false);
        acc2 = __builtin_amdgcn_wmma_f32_16x16x4_f32(false, a, false, b2, (short)0, acc2, false, false);
        acc3 = __builtin_amdgcn_wmma_f32_16x16x4_f32(false, a, false, b3, (short)0, acc3, false, false);
    }

    const int row0 = (lane >> 4) << 3;
    const int col  = lane & 15;
    const float c0 = cvec[cbase + col];
    const float c1 = cvec[cbase + 16 + col];
    const float c2 = cvec[cbase + 32 + col];
    const float c3 = cvec[cbase + 48 + col];
#pragma unroll
    for (int r = 0; r < 8; ++r) {
        float* o = qk + (size_t)(rbase + row0 + r) * D_ + cbase + col;
        o[ 0] = acc0[r] + c0;
        o[16] = acc1[r] + c1;
        o[32] = acc2[r] + c2;
        o[48] = acc3[r] + c3;
    }
}

// ---------------------------------------------------------------------------
// Kernel 2 (the HBM-bound stage, x read exactly once):
// one block per (b,l); 16 waves = 16 neurons. Wave n loads x[b,n,l,:] (2 KB),
// stages it in LDS, dots it with qk[b,l,:]; shfl-reduce -> scores; softmax
// over 16 in registers; pooled output from the LDS-cached tile.
// ---------------------------------------------------------------------------
__global__ __launch_bounds__(512)
void pool_kernel(const float* __restrict__ x, const float* __restrict__ qk,
                 float* __restrict__ out) {
    const int bl = blockIdx.x;          // b*L + l
    const int b  = bl / L_;
    const int l  = bl - b * L_;

    __shared__ float xs[N_][D_];        // 32 KB tile of x for this (b,l)
    __shared__ float sc[N_];

    const int n    = threadIdx.x >> 5;  // wave id == neuron id
    const int lane = threadIdx.x & 31;

    const float* xrow = x  + (((size_t)(b * N_ + n) * L_) + l) * D_;
    const float* qrow = qk + (size_t)bl * D_;

    // Each lane owns d in [lane*16, lane*16+16): 4 x b128 loads.
    float acc = 0.f;
#pragma unroll
    for (int c = 0; c < 4; ++c) {
        const int d = lane * 16 + c * 4;
        v4f xv = *(const v4f*)(xrow + d);
        v4f qv = *(const v4f*)(qrow + d);
        *(v4f*)(&xs[n][d]) = xv;
        acc = fmaf(xv.x, qv.x, acc);
        acc = fmaf(xv.y, qv.y, acc);
        acc = fmaf(xv.z, qv.z, acc);
        acc = fmaf(xv.w, qv.w, acc);
    }
    // wave32 tree reduction
#pragma unroll
    for (int off = 16; off >= 1; off >>= 1)
        acc += __shfl_xor(acc, off, 32);
    if (lane == 0) sc[n] = acc * RSQRT_D;
    __syncthreads();

    // softmax over the 16 neuron scores (redundant per thread; trivial)
    float s[N_];
    float mx = -INFINITY;
#pragma unroll
    for (int i = 0; i < N_; ++i) { s[i] = sc[i]; mx = fmaxf(mx, s[i]); }
    float sum = 0.f;
#pragma unroll
    for (int i = 0; i < N_; ++i) { s[i] = __expf(s[i] - mx); sum += s[i]; }
    const float inv = 1.f / sum;

    // pooled output: thread t owns d = t
    const int d = threadIdx.x;
    float o = 0.f;
#pragma unroll
    for (int i = 0; i < N_; ++i)
        o = fmaf(s[i], xs[i][d], o);
    out[(size_t)bl * D_ + d] = o * inv;
}

// ---------------------------------------------------------------------------
// Launch: inputs in setup_inputs() order: x, highway, Wq, bq, Wk, bk.
// bk is mathematically irrelevant (constant-in-n shift cancels in softmax).
// Workspace layout (floats): qk[8192*512] | M[512*512] | cvec[512]  (~17.1 MB)
// ---------------------------------------------------------------------------
extern "C" void kernel_launch(void* const* d_in, const int* in_sizes, int n_in,
                              void* d_out, int out_size, void* d_ws, size_t ws_size,
                              hipStream_t stream) {
    const float* x       = (const float*)d_in[0];
    const float* highway = (const float*)d_in[1];
    const float* Wq      = (const float*)d_in[2];
    const float* bq      = (const float*)d_in[3];
    const float* Wk      = (const float*)d_in[4];
    // d_in[5] (bk) intentionally unused: cancels in softmax over n.
    float* out = (float*)d_out;

    float* ws   = (float*)d_ws;
    float* qk   = ws;                                   // 8192*512
    float* M    = ws + (size_t)(B_ * L_) * D_;          // 512*512
    float* cvec = M + (size_t)D_ * D_;                  // 512

    // Stage 0: fold Wk into the query side. 1024 tile-waves -> 256 blocks x 4 waves.
    combine_w_kernel<<<256, 128, 0, stream>>>(Wq, Wk, M);
    bias_fold_kernel<<<1, 512, 0, stream>>>(bq, Wk, cvec);

    // Stage 1: qk = highway @ M + cvec. 4096 strip-waves -> 1024 blocks x 4 waves.
    qk_gemm_kernel<<<1024, 128, 0, stream>>>(highway, M, cvec, qk);

    // Stage 2: fused scores + softmax + pooling, one pass over x.
    pool_kernel<<<B_ * L_, 512, 0, stream>>>(x, qk, out);
}